// clip_nce_2070174237311
// MI455X (gfx1250) — compile-verified
//
#include <hip/hip_runtime.h>
#include <math.h>

typedef float v2f __attribute__((ext_vector_type(2)));
typedef float v8f __attribute__((ext_vector_type(8)));

#define ENC_NEGINF 0x007FFFFFu  // order-encoding of -inf

// Order-preserving float<->uint encoding so atomicMax(uint) == float max.
__device__ __forceinline__ unsigned fenc(float f) {
    unsigned u = __float_as_uint(f);
    return (u & 0x80000000u) ? ~u : (u | 0x80000000u);
}
__device__ __forceinline__ float fdec(unsigned e) {
    unsigned u = (e & 0x80000000u) ? (e ^ 0x80000000u) : ~e;
    return __uint_as_float(u);
}

// ---------------- init: reset segment accumulators + partials pad ----------------
__global__ void k_init(unsigned* __restrict__ segmax_enc, float* __restrict__ segsum,
                       unsigned* __restrict__ Senc, float* __restrict__ pad, int V) {
    int i = blockIdx.x * blockDim.x + threadIdx.x;
    if (i < V) { segmax_enc[i] = ENC_NEGINF; segsum[i] = 0.0f; }
    if (i < 64) { pad[i] = 0.0f; }
    if (i == 0) { *Senc = ENC_NEGINF; }
}

// ---------------- prep: gather matched scores, per-label max, global shift S ----------------
__global__ __launch_bounds__(256) void k_prep(const float* __restrict__ sc,
                                              const int* __restrict__ labels,
                                              float* __restrict__ s,
                                              unsigned* __restrict__ segmax_enc,
                                              unsigned* __restrict__ Senc,
                                              int Q, int V) {
    int q = blockIdx.x * 256 + threadIdx.x;
    if (q >= Q) return;
    int lab = labels[q];
    float sv = sc[(size_t)q * V + lab];
    s[q] = sv;
    atomicMax(&segmax_enc[lab], fenc(sv));
    // wave-level max first to keep the single-address atomic count tiny
    float m = sv;
    for (int o = 16; o; o >>= 1) m = fmaxf(m, __shfl_xor(m, o, 32));
    if ((threadIdx.x & 31) == 0) atomicMax(Senc, fenc(m));
}

// ---------------- main single pass over the 256MB matrix ----------------
// Grid: Q/64 blocks of 256 threads (8 waves). Each wave owns 8 rows; columns
// streamed in 512-wide chunks with float4 (b128) loads. exp(x-S) computed once
// per element and reused for both the row sum and the column partial sum.
__global__ __launch_bounds__(256) void k_main(const float* __restrict__ sc,
                                              const int* __restrict__ labels,
                                              const float* __restrict__ s,
                                              const unsigned* __restrict__ segmax_enc,
                                              const unsigned* __restrict__ Senc,
                                              float* __restrict__ segsum,
                                              float* __restrict__ colsum_part,
                                              float* __restrict__ t2v_part,
                                              int Q, int V, float invQ) {
    __shared__ float colLDS[8][512];
    __shared__ float t2vLDS[8];

    const int tid  = threadIdx.x;
    const int wave = tid >> 5;
    const int lane = tid & 31;
    const int rowBase = blockIdx.x * 64;
    const float S = fdec(*Senc);

    // segment-sum contributions (one thread per row of this block); segmax is final.
    if (tid < 64) {
        int q = rowBase + tid;
        int lab = labels[q];
        atomicAdd(&segsum[lab], __expf(s[q] - fdec(segmax_enc[lab])));
    }

    float rowacc[8];
#pragma unroll
    for (int r = 0; r < 8; ++r) rowacc[r] = 0.0f;

    const int waveRow0 = rowBase + wave * 8;
    const int nchunk = V >> 9;  // V / 512

    for (int c = 0; c < nchunk; ++c) {
        float c0x=0,c0y=0,c0z=0,c0w=0, c1x=0,c1y=0,c1z=0,c1w=0;
        float c2x=0,c2y=0,c2z=0,c2w=0, c3x=0,c3y=0,c3z=0,c3w=0;
        const int cbase = c * 512 + lane * 4;
#pragma unroll
        for (int r = 0; r < 8; ++r) {
            const float4* p = (const float4*)(sc + (size_t)(waveRow0 + r) * V + cbase);
            float4 v0 = p[0];
            float4 v1 = p[32];   // +128 floats
            float4 v2 = p[64];   // +256 floats
            float4 v3 = p[96];   // +384 floats
            v0.x = __expf(v0.x - S); v0.y = __expf(v0.y - S); v0.z = __expf(v0.z - S); v0.w = __expf(v0.w - S);
            v1.x = __expf(v1.x - S); v1.y = __expf(v1.y - S); v1.z = __expf(v1.z - S); v1.w = __expf(v1.w - S);
            v2.x = __expf(v2.x - S); v2.y = __expf(v2.y - S); v2.z = __expf(v2.z - S); v2.w = __expf(v2.w - S);
            v3.x = __expf(v3.x - S); v3.y = __expf(v3.y - S); v3.z = __expf(v3.z - S); v3.w = __expf(v3.w - S);
            rowacc[r] += ((v0.x + v0.y) + (v0.z + v0.w)) + ((v1.x + v1.y) + (v1.z + v1.w))
                       + ((v2.x + v2.y) + (v2.z + v2.w)) + ((v3.x + v3.y) + (v3.z + v3.w));
            c0x += v0.x; c0y += v0.y; c0z += v0.z; c0w += v0.w;
            c1x += v1.x; c1y += v1.y; c1z += v1.z; c1w += v1.w;
            c2x += v2.x; c2y += v2.y; c2z += v2.z; c2w += v2.w;
            c3x += v3.x; c3y += v3.y; c3z += v3.z; c3w += v3.w;
        }
        // per-wave column partials -> LDS (b128 stores), then deterministic block reduce
        *(float4*)&colLDS[wave][0 * 128 + lane * 4] = make_float4(c0x, c0y, c0z, c0w);
        *(float4*)&colLDS[wave][1 * 128 + lane * 4] = make_float4(c1x, c1y, c1z, c1w);
        *(float4*)&colLDS[wave][2 * 128 + lane * 4] = make_float4(c2x, c2y, c2z, c2w);
        *(float4*)&colLDS[wave][3 * 128 + lane * 4] = make_float4(c3x, c3y, c3z, c3w);
        __syncthreads();
        {
            int cc = tid * 2;
            float a0 = 0.0f, a1 = 0.0f;
#pragma unroll
            for (int w = 0; w < 8; ++w) { a0 += colLDS[w][cc]; a1 += colLDS[w][cc + 1]; }
            float2 st = make_float2(a0, a1);
            *(float2*)&colsum_part[(size_t)blockIdx.x * V + c * 512 + cc] = st;
        }
        __syncthreads();
    }

    // finish rows: cross-lane reduce, accumulate t2v terms (pre-scaled by 1/Q)
    float tw = 0.0f;
#pragma unroll
    for (int r = 0; r < 8; ++r) {
        float v = rowacc[r];
        for (int o = 16; o; o >>= 1) v += __shfl_xor(v, o, 32);
        if (lane == 0) tw += S + __logf(v) - s[waveRow0 + r];
    }
    if (lane == 0) t2vLDS[wave] = tw;
    __syncthreads();
    if (tid == 0) {
        float t = 0.0f;
        for (int w = 0; w < 8; ++w) t += t2vLDS[w];
        t2v_part[blockIdx.x] = t * invQ;
    }
}

// ---------------- reduce column partials -> per-block v2t partials (pre-scaled by 1/V) ----------------
// 64 blocks x 256 threads: each block owns 64 columns; each column's NBLK partials
// are split across 4 thread-groups, merged through LDS. Fully deterministic.
__global__ __launch_bounds__(256) void k_reduce(const float* __restrict__ colsum_part,
                                                const unsigned* __restrict__ segmax_enc,
                                                const float* __restrict__ segsum,
                                                const unsigned* __restrict__ Senc,
                                                float* __restrict__ v2t_part,
                                                int NBLK, int V, float invV) {
    __shared__ float red[256];
    const int t = threadIdx.x;
    const int cloc  = t & 63;
    const int group = t >> 6;           // 0..3
    const int c = blockIdx.x * 64 + cloc;
    const int per = NBLK >> 2;
    const float S = fdec(*Senc);

    float tot = 0.0f;
    const int p0 = group * per;
    for (int p = p0; p < p0 + per; ++p) tot += colsum_part[(size_t)p * V + c];
    red[t] = tot;
    __syncthreads();

    float term = 0.0f;
    if (t < 64) {
        float cs = (red[t] + red[t + 64]) + (red[t + 128] + red[t + 192]);
        float v2t_den = S + __logf(cs);
        float v2t_nom = fdec(segmax_enc[c]) + __logf(segsum[c]);
        term = v2t_den - v2t_nom;
    }
    __syncthreads();
    red[t] = (t < 64) ? term : 0.0f;
    __syncthreads();
    for (int s2 = 32; s2 > 0; s2 >>= 1) {
        if (t < s2) red[t] += red[t + s2];
        __syncthreads();
    }
    if (t == 0) v2t_part[blockIdx.x] = red[0] * invV;
}

// ---------------- final: WMMA (f32 16x16x4) reduction of all partials -> scalar ----------------
// One full wave32 (EXEC all ones). Partials are contiguous, pre-scaled and zero-padded,
// so loads are unconditional and all issued before the WMMA chain. B = ones so
// D[m][n] = sum_k A[m][k] + C[m][n]; chained C accumulates across iterations.
__global__ __launch_bounds__(32) void k_final(const float* __restrict__ partials,
                                              float* __restrict__ out, int nvals) {
    const int lane = threadIdx.x;
    const int ml = lane & 15;
    const int koff = (lane >> 4) * 2;
    int iters = (nvals + 63) / 64;
    if (iters > 8) iters = 8;

    v2f av[8];
#pragma unroll 8
    for (int i = 0; i < iters; ++i) {
        int i0 = i * 64 + ml * 4 + koff;
        av[i].x = partials[i0];
        av[i].y = partials[i0 + 1];
    }

    v8f c = (v8f){0.f, 0.f, 0.f, 0.f, 0.f, 0.f, 0.f, 0.f};
    v2f b; b.x = 1.0f; b.y = 1.0f;
#pragma unroll 8
    for (int i = 0; i < iters; ++i) {
        c = __builtin_amdgcn_wmma_f32_16x16x4_f32(false, av[i], false, b,
                                                  (short)0, c, false, false);
    }
    float t = ((c[0] + c[1]) + (c[2] + c[3])) + ((c[4] + c[5]) + (c[6] + c[7]));
    t += __shfl_xor(t, 16, 32);  // combine M=0..7 half with M=8..15 half
    if (lane == 0) out[0] = t;
}

extern "C" void kernel_launch(void* const* d_in, const int* in_sizes, int n_in,
                              void* d_out, int out_size, void* d_ws, size_t ws_size,
                              hipStream_t stream) {
    const float* sc     = (const float*)d_in[0];
    const int*   labels = (const int*)d_in[1];
    const int Q = in_sizes[1];
    const int V = in_sizes[0] / Q;     // 16384 x 4096
    const int NBLK = Q / 64;           // 256 main blocks
    const int NRED = V / 64;           // 64 reduce blocks -> 64 v2t partials
    const int NVALS = NBLK + NRED;     // 320 = exactly 5 WMMA tiles

    // workspace carve (all 4-byte types, base is aligned)
    float*    s           = (float*)d_ws;                    // Q
    unsigned* segmax_enc  = (unsigned*)(s + Q);              // V
    float*    segsum      = (float*)(segmax_enc + V);        // V
    unsigned* Senc        = (unsigned*)(segsum + V);         // 1 (padded to 64)
    float*    partials    = (float*)(Senc + 64);             // NVALS + 64 pad (padded to 512)
    float*    t2v_part    = partials;                        // [0, NBLK)
    float*    v2t_part    = partials + NBLK;                 // [NBLK, NVALS)
    float*    colsum_part = partials + 512;                  // NBLK * V

    k_init  <<<(V + 255) / 256, 256, 0, stream>>>(segmax_enc, segsum, Senc,
                                                  partials + NVALS, V);
    k_prep  <<<(Q + 255) / 256, 256, 0, stream>>>(sc, labels, s, segmax_enc, Senc, Q, V);
    k_main  <<<NBLK, 256, 0, stream>>>(sc, labels, s, segmax_enc, Senc,
                                       segsum, colsum_part, t2v_part, Q, V, 1.0f / (float)Q);
    k_reduce<<<NRED, 256, 0, stream>>>(colsum_part, segmax_enc, segsum, Senc,
                                       v2t_part, NBLK, V, 1.0f / (float)V);
    k_final <<<1, 32, 0, stream>>>(partials, (float*)d_out, NVALS);
}